// FCOSDetector_19061064860266
// MI455X (gfx1250) — compile-verified
//
#include <hip/hip_runtime.h>
#include <hip/hip_bf16.h>
#include <math.h>

// ---------------- constants ----------------
#define BATCH     16
#define NANCH     27280            // anchors per modality
#define NTOT      54560            // 2 modalities
#define NPAD      1008             // 1000 rounded up to 16
#define NBINS     2048
#define CANDCAP   4096
#define MAXDET    1000
#define SCORE_THR 0.05f
#define IOU_THR   0.6f
#define IMG_W     1280
#define IMG_H     1024

typedef float v2f __attribute__((ext_vector_type(2)));
typedef float v8f __attribute__((ext_vector_type(8)));

struct Ptrs { const float* p[30]; };

// ---------------- K0: init scratch ----------------
__global__ void k_init(int* hist, int* ccnt, unsigned* mask, float* cs, int* ci) {
    int stride = gridDim.x * blockDim.x;
    int t0 = blockIdx.x * blockDim.x + threadIdx.x;
    for (int t = t0; t < BATCH * NBINS; t += stride) hist[t] = 0;
    for (int t = t0; t < BATCH;         t += stride) ccnt[t] = 0;
    for (int t = t0; t < BATCH * NPAD * 32; t += stride) mask[t] = 0u;
    for (int t = t0; t < BATCH * CANDCAP; t += stride) { cs[t] = -1e30f; ci[t] = -1; }
}

// ---------------- K1: decode + score histogram ----------------
__global__ void k_decode(Ptrs P, float* sc, int* cls, float* box, int* hist) {
    int t = blockIdx.x * blockDim.x + threadIdx.x;
    if (t >= BATCH * NTOT) return;
    int b = t / NTOT;
    int g = t - b * NTOT;
    int mod = (g < NANCH) ? 0 : 1;
    int a = mod ? (g - NANCH) : g;

    const int LOFF[6] = {0, 20480, 25600, 26880, 27200, 27280};
    const int LW[5] = {160, 80, 40, 20, 10};
    const int LH[5] = {128, 64, 32, 16, 8};
    const int LS[5] = {8, 16, 32, 64, 128};
    int l = 0;
    while (a >= LOFF[l + 1]) l++;
    int w = LW[l], h = LH[l], s = LS[l];
    int idx = a - LOFF[l];
    int iy = idx / w, ix = idx - iy * w;
    float cx = (float)(ix * s + (s >> 1));
    float cy = (float)(iy * s + (s >> 1));

    const float* clsP = P.p[(mod ? 15 : 0) + l];
    const float* cntP = P.p[(mod ? 20 : 5) + l];
    const float* regP = P.p[(mod ? 25 : 10) + l];
    int hw = h * w;

    size_t coff = (size_t)b * 20 * hw + idx;
    float m = clsP[coff];
    int am = 0;
    #pragma unroll
    for (int c = 1; c < 20; c++) {
        float v = clsP[coff + (size_t)c * hw];
        if (v > m) { m = v; am = c; }
    }
    float cnt = cntP[(size_t)b * hw + idx];
    size_t roff = (size_t)b * 4 * hw + idx;
    float r0 = regP[roff], r1 = regP[roff + hw], r2 = regP[roff + 2 * (size_t)hw], r3 = regP[roff + 3 * (size_t)hw];

    // sqrt(sigmoid(m)*sigmoid(c)) == rsqrt((1+e^-m)(1+e^-c)); max(sigmoid)==sigmoid(max)
    float score = __builtin_amdgcn_rsqf((1.f + __expf(-m)) * (1.f + __expf(-cnt)));
    sc[t]  = score;
    cls[t] = am + 1;
    float* bp = box + (size_t)t * 4;
    bp[0] = cx - r0; bp[1] = cy - r1; bp[2] = cx + r2; bp[3] = cy + r3;

    int bin = (int)(score * (float)NBINS);
    bin = min(max(bin, 0), NBINS - 1);
    atomicAdd(&hist[b * NBINS + bin], 1);
}

// ---------------- K2: per-batch threshold bin ----------------
__global__ void k_thresh(const int* hist, int* tbin) {
    int b = blockIdx.x;
    if (threadIdx.x == 0) {
        int acc = 0, T = 0;
        for (int i = NBINS - 1; i >= 0; i--) {
            acc += hist[b * NBINS + i];
            if (acc >= MAXDET) { T = i; break; }
        }
        tbin[b] = T;
    }
}

// ---------------- K3: compact candidates >= threshold bin ----------------
__global__ void k_compact(const float* sc, const int* tbin, int* ccnt, float* cs, int* ci) {
    int t = blockIdx.x * blockDim.x + threadIdx.x;
    if (t >= BATCH * NTOT) return;
    int b = t / NTOT;
    int g = t - b * NTOT;
    float s = sc[t];
    int bin = (int)(s * (float)NBINS);
    bin = min(max(bin, 0), NBINS - 1);
    if (bin >= tbin[b]) {
        int slot = atomicAdd(&ccnt[b], 1);
        if (slot < CANDCAP) {
            cs[b * CANDCAP + slot] = s;
            ci[b * CANDCAP + slot] = g;
        }
    }
}

// ---------------- K4: bitonic sort (desc) + gather top-1000 + offsets ----------------
__global__ void k_sort(const float* cs, const int* ci, const int* cls, const float* box,
                       float* topS, float* topC, float* topB, float* offb, float* areaA) {
    __shared__ float ss[CANDCAP];
    __shared__ int   si[CANDCAP];
    __shared__ int   smax;
    int b = blockIdx.x, tid = threadIdx.x;
    for (int p = tid; p < CANDCAP; p += blockDim.x) { ss[p] = cs[b * CANDCAP + p]; si[p] = ci[b * CANDCAP + p]; }
    if (tid == 0) smax = 0;
    __syncthreads();

    for (int k = 2; k <= CANDCAP; k <<= 1) {
        for (int j = k >> 1; j > 0; j >>= 1) {
            for (int p = tid; p < CANDCAP; p += blockDim.x) {
                int q = p ^ j;
                if (q > p) {
                    bool dir = ((p & k) == 0);              // descending overall
                    if ((ss[p] < ss[q]) == dir) {
                        float tf = ss[p]; ss[p] = ss[q]; ss[q] = tf;
                        int   ti = si[p]; si[p] = si[q]; si[q] = ti;
                    }
                }
            }
            __syncthreads();
        }
    }

    // gather top-1000 (rest padded)
    for (int p = tid; p < NPAD; p += blockDim.x) {
        float s_ = 0.f, c_ = 0.f, b0 = 0.f, b1 = 0.f, b2 = 0.f, b3 = 0.f;
        if (p < MAXDET && si[p] >= 0) {
            int id = si[p];
            s_ = ss[p];
            c_ = (float)cls[(size_t)b * NTOT + id];
            const float* bp = box + ((size_t)b * NTOT + id) * 4;
            b0 = bp[0]; b1 = bp[1]; b2 = bp[2]; b3 = bp[3];
        }
        size_t o = (size_t)b * NPAD + p;
        topS[o] = s_; topC[o] = c_;
        float* tb = topB + o * 4;
        tb[0] = b0; tb[1] = b1; tb[2] = b2; tb[3] = b3;
        // maxc over valid entries (coords clamped to >=0; reference includes a 0 baseline)
        if (p < MAXDET && s_ >= SCORE_THR) {
            atomicMax(&smax, __float_as_int(fmaxf(fmaxf(fmaxf(b0, b1), fmaxf(b2, b3)), 0.f)));
        }
    }
    __syncthreads();
    float maxc = __int_as_float(smax);

    for (int p = tid; p < NPAD; p += blockDim.x) {
        size_t o = (size_t)b * NPAD + p;
        const float* tb = topB + o * 4;
        float off = topC[o] * (maxc + 1.f);
        float* ob = offb + o * 4;
        ob[0] = tb[0] + off; ob[1] = tb[1] + off; ob[2] = tb[2] + off; ob[3] = tb[3] + off;
        float ar = (tb[2] - tb[0] + 1.f) * (tb[3] - tb[1] + 1.f);
        areaA[o] = (p < MAXDET) ? ar : 1.f;
    }
}

// ---------------- K5: WMMA all-pairs suppression bitmask ----------------
// area[i]+area[j] per 16x16 tile via V_WMMA_F32_16X16X4_F32 (rank-2 GEMM),
// then ballot maps the WMMA C-layout directly into bitmask halfwords.
// IoU threshold via cross-multiplication: no divides in the hot loop.
__global__ void k_mask(const float* offb, const float* areaA, unsigned* mask) {
    __shared__ float sx1[NPAD], sy1[NPAD], sx2[NPAD], sy2[NPAD], sa[NPAD];
    int rt = blockIdx.x;            // row tile 0..62
    int b  = blockIdx.y;
    for (int p = threadIdx.x; p < NPAD; p += blockDim.x) {
        const float* bp = offb + ((size_t)b * NPAD + p) * 4;
        sx1[p] = bp[0]; sy1[p] = bp[1]; sx2[p] = bp[2]; sy2[p] = bp[3];
        sa[p] = areaA[(size_t)b * NPAD + p];
    }
    __syncthreads();

    int wave = threadIdx.x >> 5;
    int lane = threadIdx.x & 31;
    bool lo = lane < 16;
    int ln = lane & 15;
    int i0 = rt * 16;

    // A (16x4 f32): col0 = area_i, col1 = 1, cols 2..3 = 0
    v2f A;
    A.x = lo ? sa[i0 + ln] : 0.f;   // K0 (lanes<16) / K2 (=0)
    A.y = lo ? 1.f : 0.f;           // K1 / K3

    // row-side coords for this lane's 8 accumulator rows are r-dependent; the
    // column-side coords are fixed per ct iteration and get hoisted.
    for (int ct = wave; ct < 63; ct += 8) {   // wave-uniform loop: EXEC stays all-ones
        int j0 = ct * 16;
        int j = j0 + ln;
        float jx1 = sx1[j], jy1 = sy1[j], jx2 = sx2[j], jy2 = sy2[j];

        // B (4x16 f32): row0 = 1, row1 = area_j, rows 2..3 = 0
        v2f Bv;
        Bv.x = lo ? 1.f : 0.f;
        Bv.y = lo ? sa[j0 + ln] : 0.f;
        v8f C = {0.f, 0.f, 0.f, 0.f, 0.f, 0.f, 0.f, 0.f};
        v8f D = __builtin_amdgcn_wmma_f32_16x16x4_f32(
            false, A, false, Bv, (short)0, C, false, false);   // D[m,n] = area_i[m] + area_j[n]

        #pragma unroll
        for (int r = 0; r < 8; r++) {
            int i = i0 + r + (lo ? 0 : 8);   // WMMA C layout: vgpr r -> rows r / r+8
            float iw = fmaxf(fminf(sx2[i], jx2) - fmaxf(sx1[i], jx1) + 1.f, 0.f);
            float ih = fmaxf(fminf(sy2[i], jy2) - fmaxf(sy1[i], jy1) + 1.f, 0.f);
            float inter = iw * ih;
            float denom = D[r] - inter;
            // iou > thr  <=>  inter > thr*denom (denom > 0); denom<=0 -> iou<=0 -> false
            bool pred = (j > i) && (i < MAXDET) && (j < MAXDET) &&
                        (denom > 0.f) && (inter > IOU_THR * denom);
            unsigned bits = __builtin_amdgcn_ballot_w32(pred);
            if (lane == 0) {
                unsigned wi = (unsigned)(ct >> 1);
                unsigned sh = (unsigned)(ct & 1) * 16u;
                atomicOr(&mask[((size_t)b * NPAD + i0 + r) * 32 + wi], (bits & 0xFFFFu) << sh);
                atomicOr(&mask[((size_t)b * NPAD + i0 + r + 8) * 32 + wi], (bits >> 16) << sh);
            }
        }
    }
}

// ---------------- K6: serial greedy scan over bitmask (one wave per batch) ----------------
__global__ void k_scan(const float* topS, const unsigned* mask, int* keep) {
    __shared__ unsigned sup[32];
    int b = blockIdx.x, t = threadIdx.x;   // 32 threads
    sup[t] = 0u;
    __syncthreads();
    for (int i = 0; i < MAXDET; i++) {
        unsigned sw = sup[i >> 5];
        bool k = (topS[(size_t)b * NPAD + i] >= SCORE_THR) && !((sw >> (i & 31)) & 1u);
        if (t == 0) keep[b * MAXDET + i] = k ? 1 : 0;
        if (k) sup[t] |= mask[((size_t)b * NPAD + i) * 32 + t];
        __syncthreads();
    }
}

// ---------------- K7: clip + mask + write outputs ----------------
__global__ void k_out(const float* topS, const float* topC, const float* topB,
                      const int* keep, float* out) {
    int t = blockIdx.x * blockDim.x + threadIdx.x;
    if (t >= BATCH * MAXDET) return;
    int b = t / MAXDET;
    int k = t - b * MAXDET;
    size_t o = (size_t)b * NPAD + k;
    float kf = (float)keep[t];
    out[t] = topS[o] * kf;                            // scores
    out[BATCH * MAXDET + t] = topC[o] * kf;           // classes
    const float* tb = topB + o * 4;
    float x1 = fminf(fmaxf(tb[0], 0.f), (float)(IMG_W - 1));
    float y1 = fminf(fmaxf(tb[1], 0.f), (float)(IMG_H - 1));
    float x2 = fminf(fmaxf(tb[2], 0.f), (float)(IMG_W - 1));
    float y2 = fminf(fmaxf(tb[3], 0.f), (float)(IMG_H - 1));
    float* ob = out + 2 * BATCH * MAXDET + (size_t)t * 4;
    ob[0] = x1 * kf; ob[1] = y1 * kf; ob[2] = x2 * kf; ob[3] = y2 * kf;
    out[6 * BATCH * MAXDET + t] = kf;                 // keep
}

// ---------------- launch ----------------
extern "C" void kernel_launch(void* const* d_in, const int* in_sizes, int n_in,
                              void* d_out, int out_size, void* d_ws, size_t ws_size,
                              hipStream_t stream) {
    Ptrs P;
    for (int i = 0; i < 30; i++) P.p[i] = (const float*)d_in[i];

    char* w = (char*)d_ws;
    size_t off = 0;
    auto carve = [&](size_t bytes) { void* r = w + off; off = (off + bytes + 15) & ~(size_t)15; return r; };
    float*    sc    = (float*)    carve((size_t)BATCH * NTOT * 4);
    int*      cls   = (int*)      carve((size_t)BATCH * NTOT * 4);
    float*    box   = (float*)    carve((size_t)BATCH * NTOT * 4 * 4);
    int*      hist  = (int*)      carve((size_t)BATCH * NBINS * 4);
    int*      tbin  = (int*)      carve((size_t)BATCH * 4);
    int*      ccnt  = (int*)      carve((size_t)BATCH * 4);
    float*    cs    = (float*)    carve((size_t)BATCH * CANDCAP * 4);
    int*      ci    = (int*)      carve((size_t)BATCH * CANDCAP * 4);
    float*    topS  = (float*)    carve((size_t)BATCH * NPAD * 4);
    float*    topC  = (float*)    carve((size_t)BATCH * NPAD * 4);
    float*    topB  = (float*)    carve((size_t)BATCH * NPAD * 4 * 4);
    float*    offb  = (float*)    carve((size_t)BATCH * NPAD * 4 * 4);
    float*    areaA = (float*)    carve((size_t)BATCH * NPAD * 4);
    unsigned* maskM = (unsigned*) carve((size_t)BATCH * NPAD * 32 * 4);
    int*      keep  = (int*)      carve((size_t)BATCH * MAXDET * 4);

    k_init<<<1024, 256, 0, stream>>>(hist, ccnt, maskM, cs, ci);

    int n = BATCH * NTOT;
    k_decode<<<(n + 255) / 256, 256, 0, stream>>>(P, sc, cls, box, hist);
    k_thresh<<<BATCH, 32, 0, stream>>>(hist, tbin);
    k_compact<<<(n + 255) / 256, 256, 0, stream>>>(sc, tbin, ccnt, cs, ci);
    k_sort<<<BATCH, 1024, 0, stream>>>(cs, ci, cls, box, topS, topC, topB, offb, areaA);
    k_mask<<<dim3(63, BATCH), 256, 0, stream>>>(offb, areaA, maskM);
    k_scan<<<BATCH, 32, 0, stream>>>(topS, maskM, keep);
    k_out<<<(BATCH * MAXDET + 255) / 256, 256, 0, stream>>>(topS, topC, topB, keep, (float*)d_out);
}